// GCDEFunc_6794638262306
// MI455X (gfx1250) — compile-verified
//
#include <hip/hip_runtime.h>

typedef __attribute__((ext_vector_type(2))) float v2f;
typedef __attribute__((ext_vector_type(8))) float v8f;

#define FEAT 64

// ---------------------------------------------------------------- zeroing ---
__global__ void zero_kernel(float* __restrict__ out, int out_n,
                            int* __restrict__ deg, int n_nodes) {
    int i = blockIdx.x * blockDim.x + threadIdx.x;
    if (i < out_n) out[i] = 0.0f;
    if (i < n_nodes) deg[i] = 0;
}

// ------------------------------------------------------------- in-degree ---
__global__ void degree_kernel(const int* __restrict__ dst, int n_edges,
                              int* __restrict__ deg) {
    int e = blockIdx.x * blockDim.x + threadIdx.x;
    if (e < n_edges) atomicAdd(&deg[dst[e]], 1);
}

// ------------------------------------------------------------------ norm ---
__global__ void norm_kernel(const int* __restrict__ deg,
                            float* __restrict__ norm, int n_nodes) {
    int i = blockIdx.x * blockDim.x + threadIdx.x;
    if (i < n_nodes) {
        int d = deg[i];
        norm[i] = (d > 0) ? rsqrtf((float)d) : 0.0f;
    }
}

// ---------------------------------------------- h = (x @ W) * norm (WMMA) ---
// One wave computes a 16-row x 64-col output tile using V_WMMA_F32_16X16X4_F32.
// A (16x4 f32): lane l holds row l%16, VGPR j holds K=(l/16)*2+j.
// B (4x16 f32): lane l holds col l%16, VGPR j holds K=(l/16)*2+j.
// C/D (16x16 f32, 8 VGPRs): VGPR r, lanes 0-15 -> row r, lanes 16-31 -> row r+8.
__global__ void __launch_bounds__(256)
gemm_norm_kernel(const float* __restrict__ x, const float* __restrict__ w,
                 const float* __restrict__ norm, float* __restrict__ h,
                 int n_tiles) {
    const int wave = threadIdx.x >> 5;
    const int lane = threadIdx.x & 31;
    const int tile = blockIdx.x * 8 + wave;
    if (tile >= n_tiles) return;              // wave-uniform exit (EXEC all-1s inside)

    const int row_base = tile * 16;
    const int m   = lane & 15;                // row within tile (A) / col within tile (B)
    const int khl = (lane >> 4) * 2;          // K sub-offset for this lane half

    v8f acc0 = {}, acc1 = {}, acc2 = {}, acc3 = {};

    #pragma unroll
    for (int k0 = 0; k0 < FEAT; k0 += 4) {
        const int kk = k0 + khl;
        // A fragment: contiguous float2 from x row
        v2f a = *(const v2f*)(x + (size_t)(row_base + m) * FEAT + kk);
        // B fragments for the four 16-col tiles (column reads of W)
        v2f b0, b1, b2, b3;
        b0.x = w[(size_t)kk * FEAT +  0 + m]; b0.y = w[(size_t)(kk + 1) * FEAT +  0 + m];
        b1.x = w[(size_t)kk * FEAT + 16 + m]; b1.y = w[(size_t)(kk + 1) * FEAT + 16 + m];
        b2.x = w[(size_t)kk * FEAT + 32 + m]; b2.y = w[(size_t)(kk + 1) * FEAT + 32 + m];
        b3.x = w[(size_t)kk * FEAT + 48 + m]; b3.y = w[(size_t)(kk + 1) * FEAT + 48 + m];

        acc0 = __builtin_amdgcn_wmma_f32_16x16x4_f32(false, a, false, b0, (short)0, acc0, false, false);
        acc1 = __builtin_amdgcn_wmma_f32_16x16x4_f32(false, a, false, b1, (short)0, acc1, false, false);
        acc2 = __builtin_amdgcn_wmma_f32_16x16x4_f32(false, a, false, b2, (short)0, acc2, false, false);
        acc3 = __builtin_amdgcn_wmma_f32_16x16x4_f32(false, a, false, b3, (short)0, acc3, false, false);
    }

    // Scale each row by norm[row] and store. Lane half picks rows r / r+8.
    const int rlo = row_base + ((lane < 16) ? 0 : 8);
    const int col = lane & 15;
    #pragma unroll
    for (int r = 0; r < 8; ++r) {
        const float nm = norm[rlo + r];
        float* hp = h + (size_t)(rlo + r) * FEAT + col;
        hp[ 0] = acc0[r] * nm;
        hp[16] = acc1[r] * nm;
        hp[32] = acc2[r] * nm;
        hp[48] = acc3[r] * nm;
    }
}

// Tail path for n_nodes not divisible by 16 (not hit for N=50000, kept for safety).
__global__ void gemm_norm_tail_kernel(const float* __restrict__ x,
                                      const float* __restrict__ w,
                                      const float* __restrict__ norm,
                                      float* __restrict__ h,
                                      int row_start, int n_nodes) {
    int i = blockIdx.x * blockDim.x + threadIdx.x;
    int row = row_start + (i >> 6);
    int col = i & 63;
    if (row >= n_nodes) return;
    float acc = 0.0f;
    for (int k = 0; k < FEAT; ++k) acc += x[(size_t)row * FEAT + k] * w[(size_t)k * FEAT + col];
    h[(size_t)row * FEAT + col] = acc * norm[row];
}

// ---------------------------------------------------- edge scatter (SpMM) ---
// One wave per edge; each lane moves a float2 of the 64-feature message.
__global__ void scatter_kernel(const float* __restrict__ h,
                               const int* __restrict__ src,
                               const int* __restrict__ dst,
                               float* __restrict__ out, int n_edges) {
    long long gid = (long long)blockIdx.x * blockDim.x + threadIdx.x;
    int e = (int)(gid >> 5);
    if (e >= n_edges) return;
    int lane = (int)(gid & 31);
    int s = src[e];
    int d = dst[e];
    const float2 v = *(const float2*)(h + (size_t)s * FEAT + lane * 2);
    float* op = out + (size_t)d * FEAT + lane * 2;
    atomicAdd(op,     v.x);
    atomicAdd(op + 1, v.y);
}

// ------------------------------------- epilogue: norm + bias + softplus ----
__global__ void finalize_kernel(float* __restrict__ out,
                                const float* __restrict__ norm,
                                const float* __restrict__ bias, int out_n) {
    int i = blockIdx.x * blockDim.x + threadIdx.x;
    if (i >= out_n) return;
    float v = out[i] * norm[i >> 6] + bias[i & 63];
    // numerically stable softplus
    float r;
    if (v > 20.0f)       r = v;
    else if (v < -20.0f) r = __expf(v);
    else                 r = log1pf(__expf(v));
    out[i] = r;
}

// ----------------------------------------------------------------- launch ---
extern "C" void kernel_launch(void* const* d_in, const int* in_sizes, int n_in,
                              void* d_out, int out_size, void* d_ws, size_t ws_size,
                              hipStream_t stream) {
    // inputs: t, x, weight, bias, src, dst
    const float* x    = (const float*)d_in[1];
    const float* w    = (const float*)d_in[2];
    const float* bias = (const float*)d_in[3];
    const int*   src  = (const int*)d_in[4];
    const int*   dst  = (const int*)d_in[5];

    const int n_nodes = in_sizes[1] / FEAT;
    const int n_edges = in_sizes[4];
    float* out = (float*)d_out;

    // workspace layout: deg (int[n]) | norm (float[n]) | h (float[n*64])
    char*  ws   = (char*)d_ws;
    int*   deg  = (int*)ws;
    size_t o1   = (((size_t)n_nodes * 4) + 255) & ~(size_t)255;
    float* norm = (float*)(ws + o1);
    size_t o2   = ((o1 + (size_t)n_nodes * 4) + 255) & ~(size_t)255;
    float* h    = (float*)(ws + o2);

    const int B = 256;

    // 1) zero accumulator (d_out) and degree
    {
        int n = (out_size > n_nodes) ? out_size : n_nodes;
        zero_kernel<<<(n + B - 1) / B, B, 0, stream>>>(out, out_size, deg, n_nodes);
    }
    // 2) in-degree
    degree_kernel<<<(n_edges + B - 1) / B, B, 0, stream>>>(dst, n_edges, deg);
    // 3) norm = rsqrt(deg) or 0
    norm_kernel<<<(n_nodes + B - 1) / B, B, 0, stream>>>(deg, norm, n_nodes);
    // 4) h = (x @ W) * norm  — WMMA, 8 waves/block, one 16-row tile per wave
    {
        int n_tiles = n_nodes / 16;
        if (n_tiles > 0) {
            int blocks = (n_tiles + 7) / 8;
            gemm_norm_kernel<<<blocks, B, 0, stream>>>(x, w, norm, h, n_tiles);
        }
        int rem = n_nodes - n_tiles * 16;
        if (rem > 0) {
            int n = rem * FEAT;
            gemm_norm_tail_kernel<<<(n + B - 1) / B, B, 0, stream>>>(
                x, w, norm, h, n_tiles * 16, n_nodes);
        }
    }
    // 5) scatter-sum over edges into d_out
    {
        long long threads = (long long)n_edges * 32;
        long long blocks  = (threads + B - 1) / B;
        scatter_kernel<<<(unsigned)blocks, B, 0, stream>>>(h, src, dst, out, n_edges);
    }
    // 6) out = softplus(out * norm + bias)
    finalize_kernel<<<(out_size + B - 1) / B, B, 0, stream>>>(out, norm, bias, out_size);
}